// RoiPooling_34196529610775
// MI455X (gfx1250) — compile-verified
//
#include <hip/hip_runtime.h>
#include <hip/hip_bf16.h>
#include <stdint.h>

// RoiPooling: per-ROI crop + 7x7 bilinear resize over (1,128,128,1024) fp32.
// Memory-bound gather: uses CDNA5 async global->LDS b128 copies for the four
// bilinear corner rows, s_wait_asynccnt for completion, b128 LDS reads, and a
// coalesced b128 global store.

#define POOL   7
#define CCH    1024     // channels (innermost, contiguous)
#define WIMG   128
#define NPTS   (POOL * POOL)

__global__ __launch_bounds__(256) void roi_bilinear_kernel(
    const float* __restrict__ feat,   // [128,128,1024]
    const float* __restrict__ rois,   // [512,4] (ymin,xmin,ymax,xmax)
    float* __restrict__ out)          // [512, 49*1024]
{
    __shared__ float lds[4 * CCH];    // 4 corner rows x 4KB = 16KB

    const int tid = threadIdx.x;      // 0..255, owns channels 4*tid..4*tid+3
    const int p   = blockIdx.x;       // 0..48 sample point
    const int roi = blockIdx.y;       // 0..511
    const int py  = p / POOL;
    const int px  = p % POOL;

    // --- scalar ROI math (matches reference exactly) ---
    const float4 r = ((const float4*)rois)[roi];
    const int ymin = (int)r.x;        // astype(int32): trunc, values >= 0
    const int xmin = (int)r.y;
    const int ymax = (int)r.z;
    const int xmax = (int)r.w;

    const float h = (float)(ymax - ymin + 1);
    const float w = (float)(xmax - xmin + 1);

    const float sy = (float)py * (h / (float)POOL);
    const float sx = (float)px * (w / (float)POOL);

    const int y0 = (int)floorf(sy);
    const int x0 = (int)floorf(sx);
    const int y1 = min(y0 + 1, ymax - ymin);
    const int x1 = min(x0 + 1, xmax - xmin);

    const float fy = sy - (float)y0;
    const float fx = sx - (float)x0;

    const int gy0 = ymin + y0;
    const int gy1 = ymin + y1;
    const int gx0 = xmin + x0;
    const int gx1 = xmin + x1;

    // --- async global->LDS copies of the 4 corner channel-rows (4KB each) ---
    const uint64_t cb = (uint64_t)tid * 16u;  // this thread's 16B slice
    const uint64_t base = (uint64_t)(uintptr_t)feat;
    const uint64_t g_tl = base + ((((uint64_t)gy0 << 7) | (uint64_t)gx0) << 12) + cb;
    const uint64_t g_tr = base + ((((uint64_t)gy0 << 7) | (uint64_t)gx1) << 12) + cb;
    const uint64_t g_bl = base + ((((uint64_t)gy1 << 7) | (uint64_t)gx0) << 12) + cb;
    const uint64_t g_br = base + ((((uint64_t)gy1 << 7) | (uint64_t)gx1) << 12) + cb;

    // Low 32 bits of a flat shared-aperture address == LDS byte offset
    // (ISA 10.2 aperture rules), usable directly as the async VDST address.
    const uint32_t l0 = (uint32_t)(uintptr_t)(&lds[0]) + (uint32_t)cb;
    const uint32_t l1 = l0 + 4096u;
    const uint32_t l2 = l0 + 8192u;
    const uint32_t l3 = l0 + 12288u;

    asm volatile("global_load_async_to_lds_b128 %0, %1, off"
                 :: "v"(l0), "v"(g_tl) : "memory");
    asm volatile("global_load_async_to_lds_b128 %0, %1, off"
                 :: "v"(l1), "v"(g_tr) : "memory");
    asm volatile("global_load_async_to_lds_b128 %0, %1, off"
                 :: "v"(l2), "v"(g_bl) : "memory");
    asm volatile("global_load_async_to_lds_b128 %0, %1, off"
                 :: "v"(l3), "v"(g_br) : "memory");

    // Each thread consumes exactly the bytes it copied -> per-wave wait only,
    // no workgroup barrier needed.
    asm volatile("s_wait_asynccnt 0" ::: "memory");

    const float4* lv = (const float4*)lds;
    const float4 tl = lv[tid];
    const float4 tr = lv[256 + tid];
    const float4 bl = lv[512 + tid];
    const float4 br = lv[768 + tid];

    // Reference formula: top = tl + (tr-tl)*fx; bot = bl + (br-bl)*fx;
    //                    out = top + (bot-top)*fy
    float4 top, bot, o;
    top.x = tl.x + (tr.x - tl.x) * fx;
    top.y = tl.y + (tr.y - tl.y) * fx;
    top.z = tl.z + (tr.z - tl.z) * fx;
    top.w = tl.w + (tr.w - tl.w) * fx;
    bot.x = bl.x + (br.x - bl.x) * fx;
    bot.y = bl.y + (br.y - bl.y) * fx;
    bot.z = bl.z + (br.z - bl.z) * fx;
    bot.w = bl.w + (br.w - bl.w) * fx;
    o.x = top.x + (bot.x - top.x) * fy;
    o.y = top.y + (bot.y - top.y) * fy;
    o.z = top.z + (bot.z - top.z) * fy;
    o.w = top.w + (bot.w - top.w) * fy;

    // out[roi][py][px][c], fully coalesced b128 store
    float4* op = (float4*)(out + (((size_t)roi * NPTS + (size_t)p) << 10));
    op[tid] = o;
}

extern "C" void kernel_launch(void* const* d_in, const int* in_sizes, int n_in,
                              void* d_out, int out_size, void* d_ws, size_t ws_size,
                              hipStream_t stream) {
    const float* feat = (const float*)d_in[0];   // shared_layers (1,128,128,1024) f32
    const float* rois = (const float*)d_in[1];   // rois (512,4) f32
    float* out = (float*)d_out;                  // (512, 49*1024) f32

    (void)in_sizes; (void)n_in; (void)out_size; (void)d_ws; (void)ws_size;

    dim3 grid(NPTS, 512);   // one block per (sample point, roi)
    roi_bilinear_kernel<<<grid, 256, 0, stream>>>(feat, rois, out);
}